// NoiseLayer_60249801228333
// MI455X (gfx1250) — compile-verified
//
#include <hip/hip_runtime.h>
#include <hip/hip_bf16.h>

typedef __attribute__((ext_vector_type(2))) float v2f;
typedef __attribute__((ext_vector_type(4))) float v4f;
typedef __attribute__((ext_vector_type(8))) float v8f;

#define ROWS    16           // batch rows per block (WMMA M)
#define CHUNK   512          // samples staged per iteration
#define STRIDE  516          // 512 + 4 pad floats -> conflict-free column reads (516%64==4)
#define BLOCK   512          // 16 waves of 32

#if defined(__AMDGCN__) && __has_builtin(__builtin_amdgcn_wmma_f32_16x16x4_f32)
#define HAVE_WMMA_F32 1
#else
#define HAVE_WMMA_F32 0
#endif

// clip(s,±20) then 2*sigmoid(s)-1 == tanh(s/2); gfx1250 has native v_tanh_f32
__device__ __forceinline__ float act(float s) {
  s = fminf(fmaxf(s, -20.0f), 20.0f);
#if defined(__AMDGCN__) && __has_builtin(__builtin_amdgcn_tanhf)
  return __builtin_amdgcn_tanhf(0.5f * s);
#else
  float e = expf(-s);
  return 2.0f / (1.0f + e) - 1.0f;
#endif
}

__global__ __launch_bounds__(BLOCK) void noise_mean_wmma(
    const float* __restrict__ pub,
    const float* __restrict__ priv,
    const float* __restrict__ pb,
    const float* __restrict__ pc,
    float* __restrict__ out,
    int n_samples, float inv_n)
{
  __shared__ float tile[ROWS * STRIDE];   // 16 x 516 fp32 = ~33 KB
  __shared__ float rowsum[ROWS];

  const int tid  = threadIdx.x;
  const int lane = tid & 31;
  const int wave = tid >> 5;              // 0..15
  const int rowBase = blockIdx.x * ROWS;

  const float b = pb[0];
  const float c = pc[0];
  const int   myRow = lane & 15;          // A-fragment: lane L and 16+L both hold row M=L
  const float pt = pub[rowBase + myRow] * b;

  if (tid < ROWS) rowsum[tid] = 0.0f;

  // Coalesced stager: 4 passes, each thread one float4. Row = p*4 + tid/128, col = (tid%128)*4.
  const int ldRow = tid >> 7;             // 0..3
  const int ldCol = (tid & 127) * 4;      // 0..508

  const int kOff   = (lane >> 4) * 2;     // lanes 0-15 -> K{0,1}; lanes 16-31 -> K{2,3}
  const int waveK0 = wave * (CHUNK / 16); // 32-sample K slice per wave

#if HAVE_WMMA_F32
  v8f acc = {};                           // C/D tile; every column accumulates the row sum
  const v2f onesB = {1.0f, 1.0f};         // B = ones(4x16)
#else
  float acc = 0.0f;
#endif

  // ---- software pipeline: prefetch chunk 0 into registers ----
  v4f regs[4];
#pragma unroll
  for (int p = 0; p < 4; ++p) {
    const int r = p * 4 + ldRow;
    regs[p] = *(const v4f*)(priv + (size_t)(rowBase + r) * (size_t)n_samples
                                 + (size_t)ldCol);
  }

  for (int s0 = 0; s0 < n_samples; s0 += CHUNK) {
    __syncthreads();                      // WAR: prior compute done reading tile
#pragma unroll
    for (int p = 0; p < 4; ++p) {         // drain prefetch registers into LDS
      const int r = p * 4 + ldRow;
      *(v4f*)(tile + r * STRIDE + ldCol) = regs[p];
    }
    __syncthreads();                      // tile ready

    // issue next chunk's global loads NOW; they complete under this chunk's compute
    const int s1 = s0 + CHUNK;            // wave-uniform guard -> scalar branch, EXEC stays ~0xffffffff
    if (s1 < n_samples) {
#pragma unroll
      for (int p = 0; p < 4; ++p) {
        const int r = p * 4 + ldRow;
        regs[p] = *(const v4f*)(priv + (size_t)(rowBase + r) * (size_t)n_samples
                                     + (size_t)(s1 + ldCol));
      }
    }

#pragma unroll
    for (int k = 0; k < CHUNK / 16; k += 4) {
      const float* src = tile + myRow * STRIDE + waveK0 + k + kOff;
      v2f a;
      a.x = act(pt + src[0] * c);
      a.y = act(pt + src[1] * c);
#if HAVE_WMMA_F32
      // D[m][n] = C[m][n] + sum_k A[m][k] : reduction-by-matmul with ones
      acc = __builtin_amdgcn_wmma_f32_16x16x4_f32(false, a, false, onesB,
                                                  (short)0, acc, false, false);
#else
      acc += a.x + a.y;
#endif
    }
  }

#if HAVE_WMMA_F32
  // D layout: VGPR i, lane 0 -> (M=i, N=0); lane 16 -> (M=8+i, N=0)
  if (lane == 0 || lane == 16) {
    const int mbase = (lane >> 4) * 8;
#pragma unroll
    for (int i = 0; i < 8; ++i) atomicAdd(&rowsum[mbase + i], acc[i]);  // ds_add_f32
  }
#else
  atomicAdd(&rowsum[myRow], acc);
#endif

  __syncthreads();
  if (tid < ROWS) out[rowBase + tid] = rowsum[tid] * inv_n;
}

extern "C" void kernel_launch(void* const* d_in, const int* in_sizes, int n_in,
                              void* d_out, int out_size, void* d_ws, size_t ws_size,
                              hipStream_t stream) {
  const float* pub  = (const float*)d_in[0];   // (batch,)
  const float* priv = (const float*)d_in[1];   // (batch, n_samples, 1)
  const float* pb   = (const float*)d_in[2];   // (1,1)
  const float* pc   = (const float*)d_in[3];   // (1,1)
  float* out = (float*)d_out;                  // (batch,1,1)

  const int batch     = in_sizes[0];
  const int n_samples = in_sizes[1] / batch;
  const float inv_n   = 1.0f / (float)n_samples;

  dim3 grid(batch / ROWS);                     // 4096/16 = 256 blocks, block-exclusive output
  noise_mean_wmma<<<grid, BLOCK, 0, stream>>>(pub, priv, pb, pc, out, n_samples, inv_n);
}